// EmbeddingDropout_70592082477707
// MI455X (gfx1250) — compile-verified
//
#include <hip/hip_runtime.h>

// EmbeddingDropout gather: out[b,s,:] = weight[x[b,s],:] * mask[x[b,s]]
// Memory-bound (≈64 MB @ 23.3 TB/s ≈ 2.8 us floor; FLOPs negligible).
// CDNA5 path: Tensor Data Mover GATHER mode. Each block covers 32 tokens via
// TWO 16-row gather descriptors issued back-to-back from wave 0. TDM ops from
// one wave complete in-order, so s_wait_tensorcnt 1 == "first batch landed",
// letting batch-A store-out overlap batch-B DMA. Output uses non-temporal
// 128-bit stores so the 32 MB output stream doesn't evict the weight table
// from L2 (103 MB < 192 MB).

typedef __attribute__((ext_vector_type(4))) float        v4f;
typedef __attribute__((ext_vector_type(4))) unsigned int v4u;
typedef __attribute__((ext_vector_type(4))) int          v4i;
typedef __attribute__((ext_vector_type(8))) int          v8i;

#define ROWS_PER_BATCH   16      // rows per TDM gather descriptor (16-bit idx mode)
#define TOKENS_PER_BLOCK 32      // two pipelined batches
#define D_DIM            512     // floats per embedding row
#define D_VEC            (D_DIM / 4)
#define V_DIM            50257   // vocab size (fits in 16-bit TDM gather index)

__device__ __forceinline__ v4u make_group0(unsigned lds_off, unsigned long long wa) {
    v4u g0;
    g0[0] = 1u                      // count = 1 valid descriptor
          | (0u << 30)              // gather_index_size = 16-bit
          | (1u << 31);             // gather_mode = 1
    g0[1] = lds_off;                // lds_addr (bytes)
    g0[2] = (unsigned)(wa & 0xFFFFFFFFull);          // global_addr[31:0]
    g0[3] = (unsigned)((wa >> 32) & 0x01FFFFFFull)   // global_addr[56:32]
          | (2u << 30);                              // type = 2 (image)
    return g0;
}

__global__ __launch_bounds__(256)
void EmbeddingDropout_70592082477707_kernel(const int*   __restrict__ xp,
                                            const float* __restrict__ weightp,
                                            const float* __restrict__ maskp,
                                            float*       __restrict__ outp)
{
    __shared__ v4f   tile[2][ROWS_PER_BATCH * D_VEC];  // 2 x 32 KB double buffer
    __shared__ float smask[TOKENS_PER_BLOCK];

    const int t0  = blockIdx.x * TOKENS_PER_BLOCK;
    const int tid = threadIdx.x;

    // ---- Wave 0: fetch 32 indices, build two gather descriptors, issue both ----
    if (tid < 32) {
        const int lane = tid;
        const int idx  = xp[t0 + lane];          // one coalesced b32 load
        smask[lane] = maskp[idx];                // per-row dropout scale (gather)

        // Pack 32 row indices as 16-bit pairs: lanes 0-15 -> batch A (8 words),
        // lanes 16-31 -> batch B (8 words).
        unsigned p[16];
#pragma unroll
        for (int k = 0; k < 16; ++k) {
            unsigned lo = (unsigned)__builtin_amdgcn_readlane(idx, 2 * k)     & 0xFFFFu;
            unsigned hi = (unsigned)__builtin_amdgcn_readlane(idx, 2 * k + 1) & 0xFFFFu;
            p[k] = lo | (hi << 16);
        }

        // LDS byte offsets: generic LDS addresses carry the workgroup-relative
        // offset in addr[31:0].
        unsigned ldsA = (unsigned)(uintptr_t)&tile[0][0];
        unsigned ldsB = (unsigned)(uintptr_t)&tile[1][0];
        unsigned long long wa = (unsigned long long)(uintptr_t)weightp;

        // ---- shared D# group 1 (8 dwords) ----
        v8i g1;
        g1[0] = (int)(2u << 16);                 // workgroup_mask=0, data_size=4B
        g1[1] = (int)((unsigned)D_DIM << 16);    // tensor_dim0[15:0] = 512
        g1[2] = (int)((unsigned)V_DIM << 16);    // dim0 hi = 0, tensor_dim1[15:0] = V
        g1[3] = (int)((unsigned)D_DIM << 16);    // dim1 hi = 0, tile_dim0 = 512
        g1[4] = ROWS_PER_BATCH;                  // tile_dim1 = #valid gather indices
        g1[5] = D_DIM;                           // tensor_dim0_stride = 512 elements
        g1[6] = 0;                               // dim1_stride ignored in gather mode
        g1[7] = 0;

        v4u g0A = make_group0(ldsA, wa);
        v4u g0B = make_group0(ldsB, wa);
        v4i g2A = { (int)p[0],  (int)p[1],  (int)p[2],  (int)p[3]  };
        v4i g3A = { (int)p[4],  (int)p[5],  (int)p[6],  (int)p[7]  };
        v4i g2B = { (int)p[8],  (int)p[9],  (int)p[10], (int)p[11] };
        v4i g3B = { (int)p[12], (int)p[13], (int)p[14], (int)p[15] };
        v8i g4  = { 0, 0, 0, 0, 0, 0, 0, 0 };    // clang-23 6-arg extra block

        __builtin_amdgcn_tensor_load_to_lds(g0A, g1, g2A, g3A, g4, /*cpol=*/0);
        __builtin_amdgcn_tensor_load_to_lds(g0B, g1, g2B, g3B, g4, /*cpol=*/0);

        // TDM ops from one wave complete in order: <=1 outstanding means the
        // FIRST gather (batch A) has fully landed in LDS; B still in flight.
        __builtin_amdgcn_s_wait_tensorcnt(1);
    }
    __syncthreads();

    v4f* out4 = (v4f*)outp;

    // ---- Batch A: scale + NT stream-out, overlapped with batch B's DMA ----
#pragma unroll
    for (int j = 0; j < 8; ++j) {
        int q = tid + j * 256;                   // [0, 2048) float4 chunks
        int r = q >> 7;                          // token row within batch
        int c = q & (D_VEC - 1);                 // float4 column
        v4f v = tile[0][(r << 7) + c];           // ds_load_b128
        v4f s = v * smask[r];
        __builtin_nontemporal_store(s, &out4[(size_t)(t0 + r) * D_VEC + c]);
    }

    // ---- Drain batch B (counter lives in wave 0, the issuing wave) ----
    if (tid < 32) {
        __builtin_amdgcn_s_wait_tensorcnt(0);
    }
    __syncthreads();

#pragma unroll
    for (int j = 0; j < 8; ++j) {
        int q = tid + j * 256;
        int r = q >> 7;
        int c = q & (D_VEC - 1);
        v4f v = tile[1][(r << 7) + c];
        v4f s = v * smask[ROWS_PER_BATCH + r];
        __builtin_nontemporal_store(
            s, &out4[(size_t)(t0 + ROWS_PER_BATCH + r) * D_VEC + c]);
    }
}

extern "C" void kernel_launch(void* const* d_in, const int* in_sizes, int n_in,
                              void* d_out, int out_size, void* d_ws, size_t ws_size,
                              hipStream_t stream) {
    const int*   xp = (const int*)d_in[0];     // [B,S] token ids
    const float* wp = (const float*)d_in[1];   // [V,D] embedding table
    const float* mp = (const float*)d_in[2];   // [V,1] row dropout scale
    float*       op = (float*)d_out;           // [B,S,D]

    const int n_tok  = in_sizes[0];            // 8*2048 = 16384
    const int blocks = n_tok / TOKENS_PER_BLOCK;

    EmbeddingDropout_70592082477707_kernel<<<blocks, 256, 0, stream>>>(xp, wp, mp, op);
}